// WassersteinLoss_70196945486185
// MI455X (gfx1250) — compile-verified
//
#include <hip/hip_runtime.h>
#include <hip/hip_bf16.h>

#define N 8192
#define D 64
#define INV_EPS 10.0f   // 1/0.1
#define EPS 0.1f

typedef float v2f __attribute__((ext_vector_type(2)));
typedef float v8f __attribute__((ext_vector_type(8)));

// ---------------------------------------------------------------------------
// Kernel 1: row norms of P and Q, and u = 1/D
// ---------------------------------------------------------------------------
__global__ void prep_kernel(const float* __restrict__ P, const float* __restrict__ Q,
                            float* __restrict__ pp, float* __restrict__ qq,
                            float* __restrict__ u) {
    int i = blockIdx.x * blockDim.x + threadIdx.x;
    if (i < N) {
        float sp = 0.0f, sq = 0.0f;
        const float* pr = P + (size_t)i * D;
        const float* qr = Q + (size_t)i * D;
#pragma unroll 8
        for (int k = 0; k < D; ++k) {
            float a = pr[k]; sp = fmaf(a, a, sp);
            float b = qr[k]; sq = fmaf(b, b, sq);
        }
        pp[i] = sp;
        qq[i] = sq;
        u[i] = 1.0f / (float)D;
    }
}

// ---------------------------------------------------------------------------
// Kernel 2: K = exp(-cdist(P,Q)/eps) via V_WMMA_F32_16X16X4_F32.
// Block = 256 threads = 8 waves; wave w computes the 16x16 tile at
// rows [16*blockIdx.y, +16), cols [16*(8*blockIdx.x + w), +16).
//
// A fragment (16x4 f32, 2 VGPRs): lane l (l<16), half h: holds
//   P[row0+l][kb*4 + 2h + {0,1}]
// B fragment (4x16 f32, 2 VGPRs) mirrors A with B[k][n] = Q[col0+n][k]:
//   Q[col0+l][kb*4 + 2h + {0,1}]
// D layout (8 VGPRs): acc[r] = G[row0 + r + 8h][col0 + (lane&15)]
// ---------------------------------------------------------------------------
__global__ void gemm_expK_kernel(const float* __restrict__ P, const float* __restrict__ Q,
                                 const float* __restrict__ pp, const float* __restrict__ qq,
                                 float* __restrict__ Kmat) {
    const int lane = threadIdx.x & 31;
    const int wave = threadIdx.x >> 5;
    const int row0 = blockIdx.y * 16;
    const int col0 = (blockIdx.x * 8 + wave) * 16;
    const int h = lane >> 4;
    const int l = lane & 15;

    const float* prow = P + (size_t)(row0 + l) * D;
    const float* qrow = Q + (size_t)(col0 + l) * D;

    v8f acc = {0.f, 0.f, 0.f, 0.f, 0.f, 0.f, 0.f, 0.f};
#pragma unroll
    for (int kb = 0; kb < D / 4; ++kb) {
        const int kidx = kb * 4 + 2 * h;
        v2f a = *(const v2f*)(prow + kidx);
        v2f b = *(const v2f*)(qrow + kidx);
        acc = __builtin_amdgcn_wmma_f32_16x16x4_f32(
            /*neg_a=*/false, a, /*neg_b=*/false, b,
            /*c_mod=*/(short)0, acc, /*reuse_a=*/false, /*reuse_b=*/false);
    }

    const float qqc = qq[col0 + l];
#pragma unroll
    for (int r = 0; r < 8; ++r) {
        const int row = row0 + r + 8 * h;
        float S = pp[row] + qqc - 2.0f * acc[r];
        float c = __builtin_sqrtf(fmaxf(S, 0.0f));
        Kmat[(size_t)row * N + col0 + l] = __expf(-c * INV_EPS);
    }
}

// ---------------------------------------------------------------------------
// Kernel 3: out[i] = 1 / sum_j K[i][j] * x[j]    (one block per row)
// ---------------------------------------------------------------------------
__global__ void matvec_recip_kernel(const float* __restrict__ Kmat,
                                    const float* __restrict__ x,
                                    float* __restrict__ out) {
    __shared__ float red[256];
    const int i = blockIdx.x;
    const float* row = Kmat + (size_t)i * N;
    float s = 0.0f;
    for (int j = threadIdx.x * 4; j < N; j += 256 * 4) {
        float4 k4 = *(const float4*)(row + j);
        float4 x4 = *(const float4*)(x + j);
        s = fmaf(k4.x, x4.x, s);
        s = fmaf(k4.y, x4.y, s);
        s = fmaf(k4.z, x4.z, s);
        s = fmaf(k4.w, x4.w, s);
    }
    red[threadIdx.x] = s;
    __syncthreads();
    for (int off = 128; off > 0; off >>= 1) {
        if ((int)threadIdx.x < off) red[threadIdx.x] += red[threadIdx.x + off];
        __syncthreads();
    }
    if (threadIdx.x == 0) out[i] = 1.0f / red[0];
}

// ---------------------------------------------------------------------------
// Kernel 4: result += u[i] * sum_j K[i][j] * v[j] * C[i][j],
// recovering C = -eps * ln(K). One block per row, one atomicAdd per block.
// ---------------------------------------------------------------------------
__global__ void final_sum_kernel(const float* __restrict__ Kmat,
                                 const float* __restrict__ u,
                                 const float* __restrict__ v,
                                 float* __restrict__ out) {
    __shared__ float red[256];
    const int i = blockIdx.x;
    const float* row = Kmat + (size_t)i * N;
    float s = 0.0f;
    for (int j = threadIdx.x * 4; j < N; j += 256 * 4) {
        float4 k4 = *(const float4*)(row + j);
        float4 v4 = *(const float4*)(v + j);
        s += k4.x * v4.x * (-EPS * __logf(k4.x));
        s += k4.y * v4.y * (-EPS * __logf(k4.y));
        s += k4.z * v4.z * (-EPS * __logf(k4.z));
        s += k4.w * v4.w * (-EPS * __logf(k4.w));
    }
    red[threadIdx.x] = s;
    __syncthreads();
    for (int off = 128; off > 0; off >>= 1) {
        if ((int)threadIdx.x < off) red[threadIdx.x] += red[threadIdx.x + off];
        __syncthreads();
    }
    if (threadIdx.x == 0) atomicAdd(out, u[i] * red[0]);
}

// ---------------------------------------------------------------------------
extern "C" void kernel_launch(void* const* d_in, const int* in_sizes, int n_in,
                              void* d_out, int out_size, void* d_ws, size_t ws_size,
                              hipStream_t stream) {
    (void)in_sizes; (void)n_in; (void)out_size; (void)ws_size;
    const float* P = (const float*)d_in[0];
    const float* Q = (const float*)d_in[1];

    float* ws  = (float*)d_ws;
    float* Kmat = ws;                       // N*N floats (256 MiB)
    float* u    = Kmat + (size_t)N * N;     // N
    float* v    = u + N;                    // N
    float* pp   = v + N;                    // N
    float* qq   = pp + N;                   // N

    // 1) row norms + u init
    prep_kernel<<<(N + 255) / 256, 256, 0, stream>>>(P, Q, pp, qq, u);

    // 2) K = exp(-cdist/eps) via WMMA f32
    dim3 grid(N / 128, N / 16);             // 64 x 512 blocks, 8 waves each
    gemm_expK_kernel<<<grid, 256, 0, stream>>>(P, Q, pp, qq, Kmat);

    // 3) Sinkhorn: v = 1/(K u); u = 1/(K v)  x5
    for (int it = 0; it < 5; ++it) {
        matvec_recip_kernel<<<N, 256, 0, stream>>>(Kmat, u, v);
        matvec_recip_kernel<<<N, 256, 0, stream>>>(Kmat, v, u);
    }

    // 4) result = sum(u_i K_ij v_j C_ij)
    hipMemsetAsync(d_out, 0, sizeof(float), stream);
    final_sum_kernel<<<N, 256, 0, stream>>>(Kmat, u, v, (float*)d_out);
}